// RetinaNet_69398081568967
// MI455X (gfx1250) — compile-verified
//
#include <hip/hip_runtime.h>
#include <stdint.h>

#define TOPK       1000
#define CAND_CAP   4096
#define NBINS      65536
#define PAD        1024      // per-class padded candidate stride
#define SCORE_THR  0.05f
#define IOU_THR    0.5f

#if defined(__HIP_DEVICE_COMPILE__) && defined(__gfx1250__)
#  if __has_builtin(__builtin_amdgcn_tensor_load_to_lds) && __has_builtin(__builtin_amdgcn_s_wait_tensorcnt)
#    define HAVE_TDM 1
#  endif
#endif
#ifndef HAVE_TDM
#  define HAVE_TDM 0
#endif

typedef unsigned int v4u __attribute__((ext_vector_type(4)));
typedef int          v4i __attribute__((ext_vector_type(4)));
typedef int          v8i __attribute__((ext_vector_type(8)));

#if HAVE_TDM
// 1D contiguous DMA: global -> LDS via the Tensor Data Mover.
//   valid_elems : tensor_dim0 -> elements past this arrive as ZERO (TDM OOB rule)
//   tile_elems  : tile_dim0   -> elements written to LDS (<= 65535)
//   dsize_code  : 0=1B, 1=2B, 2=4B, 3=8B
// Caller must be a single wave; caller does s_wait_tensorcnt + barrier.
__device__ __forceinline__ void tdm_load_1d(void* lds_dst, const void* gsrc,
                                            unsigned int valid_elems,
                                            unsigned int tile_elems,
                                            unsigned int dsize_code) {
    unsigned long long gaddr = (unsigned long long)(uintptr_t)gsrc;
    v4u g0 = {0u, 0u, 0u, 0u};
    g0[0] = 1u;                                                       // count=1, user descriptor
    g0[1] = (unsigned int)(uintptr_t)lds_dst;                         // low 32 bits = LDS offset
    g0[2] = (unsigned int)(gaddr & 0xFFFFFFFFu);                      // global_addr[31:0]
    g0[3] = (unsigned int)((gaddr >> 32) & 0x1FFFFFFu) | (2u << 30);  // addr[56:32] | type=2
    v8i g1 = {0, 0, 0, 0, 0, 0, 0, 0};
    g1[0] = (int)(dsize_code << 16);                                  // data_size
    g1[1] = (int)((valid_elems & 0xFFFFu) << 16);                     // tensor_dim0[15:0]
    g1[2] = (int)(((valid_elems >> 16) & 0xFFFFu) | (1u << 16));      // tensor_dim0[31:16] | tensor_dim1=1
    g1[3] = (int)((tile_elems & 0xFFFFu) << 16);                      // tile_dim0
    g1[4] = 1;                                                        // tile_dim1=1, tile_dim2=0
    g1[5] = (int)tile_elems;                                          // tensor_dim0_stride[31:0]
    v4i z4 = {0, 0, 0, 0};
#if __clang_major__ >= 23
    v8i z8 = {0, 0, 0, 0, 0, 0, 0, 0};
    __builtin_amdgcn_tensor_load_to_lds(g0, g1, z4, z4, z8, 0);
#else
    __builtin_amdgcn_tensor_load_to_lds(g0, g1, z4, z4, 0);
#endif
}
#endif

// ---------------------------------------------------------------- K0: zero
__global__ void k_zero_u32(unsigned int* __restrict__ p, long long n) {
    long long i  = (long long)blockIdx.x * blockDim.x + threadIdx.x;
    long long st = (long long)gridDim.x * blockDim.x;
    for (; i < n; i += st) p[i] = 0u;
}

// ---------------------------------------------------------------- K1: decode + clip
__global__ void k_decode(const float* __restrict__ anchors,
                         const float* __restrict__ reg,
                         const int* __restrict__ img_h,
                         const int* __restrict__ img_w,
                         float* __restrict__ boxes, int A) {
    int a = blockIdx.x * blockDim.x + threadIdx.x;
    if (a >= A) return;
    float a0 = anchors[a * 4 + 0], a1 = anchors[a * 4 + 1];
    float a2 = anchors[a * 4 + 2], a3 = anchors[a * 4 + 3];
    float dx = reg[a * 4 + 0] * 0.1f, dy = reg[a * 4 + 1] * 0.1f;
    float dw = reg[a * 4 + 2] * 0.2f, dh = reg[a * 4 + 3] * 0.2f;
    float w  = a2 - a0, h = a3 - a1;
    float cx = a0 + 0.5f * w, cy = a1 + 0.5f * h;
    float pcx = cx + dx * w, pcy = cy + dy * h;
    float pw  = expf(dw) * w, ph = expf(dh) * h;
    float W = (float)img_w[0], H = (float)img_h[0];
    boxes[a * 4 + 0] = fminf(fmaxf(pcx - 0.5f * pw, 0.f), W);
    boxes[a * 4 + 1] = fminf(fmaxf(pcy - 0.5f * ph, 0.f), H);
    boxes[a * 4 + 2] = fminf(fmaxf(pcx + 0.5f * pw, 0.f), W);
    boxes[a * 4 + 3] = fminf(fmaxf(pcy + 0.5f * ph, 0.f), H);
}

// ---------------------------------------------------------------- K2: per-class 16-bit key histogram
__global__ void k_hist(const float* __restrict__ scores,
                       unsigned int* __restrict__ hist,
                       long long N, int C) {
    long long i  = (long long)blockIdx.x * blockDim.x + threadIdx.x;
    long long st = (long long)gridDim.x * blockDim.x;
    for (; i < N; i += st) {
        float s = scores[i];
        if (s > SCORE_THR) {
            int c = (int)(i % C);
            unsigned int key = __float_as_uint(s) >> 16;  // monotonic for s > 0
            atomicAdd(&hist[(size_t)c * NBINS + key], 1u);
        }
    }
}

// ---------------------------------------------------------------- K3: cutoff key per class
__global__ void k_cutoff(const unsigned int* __restrict__ hist,
                         unsigned int* __restrict__ cutkey,
                         unsigned int* __restrict__ cnt) {
    int c = blockIdx.x;
    __shared__ unsigned int part[256];
    const unsigned int* h = hist + (size_t)c * NBINS;
    const int CHUNK = NBINS / 256;
    unsigned int s = 0;
    int base = threadIdx.x * CHUNK;
    for (int k = 0; k < CHUNK; ++k) s += h[base + k];
    part[threadIdx.x] = s;
    __syncthreads();
    if (threadIdx.x == 0) {
        unsigned int acc = 0, cut = 0;
        for (int t = 255; t >= 0; --t) {
            if (acc + part[t] >= (unsigned)TOPK) {
                for (int b = (t + 1) * CHUNK - 1; b >= t * CHUNK; --b) {
                    acc += h[b];
                    if (acc >= (unsigned)TOPK) { cut = (unsigned)b; break; }
                }
                break;
            }
            acc += part[t];
        }
        cutkey[c] = cut;   // cut==0 when fewer than TOPK pass the threshold
        cnt[c]    = 0u;
    }
}

// ---------------------------------------------------------------- K4: gather candidates (composite keys)
__global__ void k_gather(const float* __restrict__ scores,
                         const unsigned int* __restrict__ cutkey,
                         unsigned int* __restrict__ cnt,
                         unsigned long long* __restrict__ ckey,
                         long long N, int C) {
    long long i  = (long long)blockIdx.x * blockDim.x + threadIdx.x;
    long long st = (long long)gridDim.x * blockDim.x;
    for (; i < N; i += st) {
        __builtin_prefetch(&scores[i + st], 0, 1);  // -> global_prefetch_b8 (speculative OK)
        float s = scores[i];
        if (s > SCORE_THR) {
            int c = (int)(i % C);
            unsigned int bits = __float_as_uint(s);
            if ((bits >> 16) >= cutkey[c]) {
                unsigned int pos = atomicAdd(&cnt[c], 1u);
                if (pos < CAND_CAP) {
                    unsigned int a = (unsigned int)(i / C);
                    ckey[(size_t)c * CAND_CAP + pos] =
                        ((unsigned long long)bits << 32) | (0xFFFFFFFFu - a);
                }
            }
        }
    }
}

// ---------------------------------------------------------------- K5: bitonic sort + emit per-class tiles
__global__ void __launch_bounds__(1024)
k_sort_select(const unsigned long long* __restrict__ ckey,
              const unsigned int* __restrict__ cnt,
              const float* __restrict__ boxes,
              float* __restrict__ cscore,
              float* __restrict__ cboxes) {
    int c = blockIdx.x;
    __shared__ unsigned long long keys[CAND_CAP];   // 32 KB
    unsigned int n = cnt[c];
    if (n > CAND_CAP) n = CAND_CAP;

#if HAVE_TDM
    // TDM: tensor_dim0 = n  -> elements [n, CAND_CAP) arrive as zeros (OOB rule),
    // which is exactly the padding the descending bitonic sort needs.
    if (threadIdx.x < 32) {
        tdm_load_1d(&keys[0], ckey + (size_t)c * CAND_CAP, n, CAND_CAP, 3u);
        __builtin_amdgcn_s_wait_tensorcnt(0);
    }
    __syncthreads();
#else
    for (int i = threadIdx.x; i < CAND_CAP; i += blockDim.x)
        keys[i] = (i < (int)n) ? ckey[(size_t)c * CAND_CAP + i] : 0ull;
    __syncthreads();
#endif

    // descending bitonic sort; keys are unique -> order deterministic
    for (int k = 2; k <= CAND_CAP; k <<= 1) {
        for (int j = k >> 1; j > 0; j >>= 1) {
            for (int i = threadIdx.x; i < CAND_CAP; i += blockDim.x) {
                int l = i ^ j;
                if (l > i) {
                    unsigned long long a = keys[i], b = keys[l];
                    bool up = ((i & k) == 0);
                    if ((a < b) == up) { keys[i] = b; keys[l] = a; }
                }
            }
            __syncthreads();
        }
    }
    for (int i = threadIdx.x; i < PAD; i += blockDim.x) {
        unsigned long long kk = (i < TOPK) ? keys[i] : 0ull;
        float s = 0.f, b0 = 0.f, b1 = 0.f, b2 = 0.f, b3 = 0.f;
        if (kk != 0ull) {
            s = __uint_as_float((unsigned int)(kk >> 32));
            unsigned int a = 0xFFFFFFFFu - (unsigned int)(kk & 0xFFFFFFFFull);
            b0 = boxes[(size_t)a * 4 + 0]; b1 = boxes[(size_t)a * 4 + 1];
            b2 = boxes[(size_t)a * 4 + 2]; b3 = boxes[(size_t)a * 4 + 3];
        }
        cscore[(size_t)c * PAD + i] = s;
        size_t o = ((size_t)c * PAD + i) * 4;
        cboxes[o + 0] = b0; cboxes[o + 1] = b1;
        cboxes[o + 2] = b2; cboxes[o + 3] = b3;
    }
}

// ---------------------------------------------------------------- K6: greedy NMS (TDM tile loads into LDS)
__global__ void __launch_bounds__(1024)
k_nms(const float* __restrict__ cscore,
      const float* __restrict__ cboxes,
      float* __restrict__ out, int C) {
    int c = blockIdx.x;
    int tid = threadIdx.x;
    __shared__ float sb[PAD * 4];    // 16 KB box tile
    __shared__ float sarea[PAD];
    __shared__ float sscore[PAD];    // 4 KB score tile
    __shared__ int   skeep[PAD];

#if HAVE_TDM
    if (tid < 32) {  // one wave issues both TDM ops; in-order -> single wait covers both
        tdm_load_1d(&sb[0],     cboxes + (size_t)c * PAD * 4, PAD * 4, PAD * 4, 2u);
        tdm_load_1d(&sscore[0], cscore + (size_t)c * PAD,     PAD,     PAD,     2u);
        __builtin_amdgcn_s_wait_tensorcnt(0);
    }
    __syncthreads();
#else
    for (int i = tid; i < PAD * 4; i += blockDim.x)
        sb[i] = cboxes[(size_t)c * PAD * 4 + i];
    for (int i = tid; i < PAD; i += blockDim.x)
        sscore[i] = cscore[(size_t)c * PAD + i];
    __syncthreads();
#endif

    float x1 = sb[tid * 4 + 0], y1 = sb[tid * 4 + 1];
    float x2 = sb[tid * 4 + 2], y2 = sb[tid * 4 + 3];
    sarea[tid] = fmaxf(x2 - x1, 0.f) * fmaxf(y2 - y1, 0.f);
    skeep[tid] = (tid < TOPK) && (sscore[tid] > SCORE_THR);
    __syncthreads();

    for (int i = 0; i < TOPK; ++i) {
        if (skeep[i] && tid > i && tid < TOPK && skeep[tid]) {
            float ix1 = fmaxf(x1, sb[i * 4 + 0]);
            float iy1 = fmaxf(y1, sb[i * 4 + 1]);
            float ix2 = fminf(x2, sb[i * 4 + 2]);
            float iy2 = fminf(y2, sb[i * 4 + 3]);
            float inter = fmaxf(ix2 - ix1, 0.f) * fmaxf(iy2 - iy1, 0.f);
            float uni   = sarea[tid] + sarea[i] - inter;
            if (inter / fmaxf(uni, 1e-8f) > IOU_THR) skeep[tid] = 0;
        }
        __syncthreads();
    }

    if (tid < TOPK) {
        bool kp = skeep[tid] != 0;
        size_t o     = (size_t)c * TOPK + tid;
        size_t total = (size_t)C * TOPK;
        out[o]         = kp ? sscore[tid] : 0.f;
        out[total + o] = kp ? (float)c : -1.0f;
        size_t ob = total * 2 + o * 4;
        out[ob + 0] = kp ? x1 : 0.f;
        out[ob + 1] = kp ? y1 : 0.f;
        out[ob + 2] = kp ? x2 : 0.f;
        out[ob + 3] = kp ? y2 : 0.f;
    }
}

// ---------------------------------------------------------------- launcher
extern "C" void kernel_launch(void* const* d_in, const int* in_sizes, int n_in,
                              void* d_out, int out_size, void* d_ws, size_t ws_size,
                              hipStream_t stream) {
    const float* cls     = (const float*)d_in[0];  // [1, A, C]
    const float* reg     = (const float*)d_in[1];  // [1, A, 4]
    const float* anchors = (const float*)d_in[2];  // [A, 4]
    const int*   img_h   = (const int*)d_in[3];
    const int*   img_w   = (const int*)d_in[4];
    float* out = (float*)d_out;

    int A = in_sizes[1] / 4;
    int C = (A > 0) ? in_sizes[0] / A : 0;
    long long N = (long long)A * C;

    char* ws = (char*)d_ws;
    size_t off = 0;
    auto alloc = [&](size_t bytes) -> void* {
        off = (off + 255) & ~(size_t)255;
        void* p = ws + off;
        off += bytes;
        return p;
    };
    unsigned int*       hist   = (unsigned int*)alloc((size_t)C * NBINS * sizeof(unsigned int));
    float*              boxes  = (float*)alloc((size_t)A * 4 * sizeof(float));
    unsigned int*       cutkey = (unsigned int*)alloc((size_t)C * sizeof(unsigned int));
    unsigned int*       cnt    = (unsigned int*)alloc((size_t)C * sizeof(unsigned int));
    unsigned long long* ckey   = (unsigned long long*)alloc((size_t)C * CAND_CAP * sizeof(unsigned long long));
    float*              cscore = (float*)alloc((size_t)C * PAD * sizeof(float));
    float*              cboxes = (float*)alloc((size_t)C * PAD * 4 * sizeof(float));

    long long histN = (long long)C * NBINS;
    k_zero_u32<<<1024, 256, 0, stream>>>(hist, histN);
    k_decode<<<(A + 255) / 256, 256, 0, stream>>>(anchors, reg, img_h, img_w, boxes, A);
    k_hist<<<8192, 256, 0, stream>>>(cls, hist, N, C);
    k_cutoff<<<C, 256, 0, stream>>>(hist, cutkey, cnt);
    k_gather<<<8192, 256, 0, stream>>>(cls, cutkey, cnt, ckey, N, C);
    k_sort_select<<<C, 1024, 0, stream>>>(ckey, cnt, boxes, cscore, cboxes);
    k_nms<<<C, 1024, 0, stream>>>(cscore, cboxes, out, C);
}